// StochPool_72559177499179
// MI455X (gfx1250) — compile-verified
//
#include <hip/hip_runtime.h>
#include <cstdint>
#include <cstddef>

typedef __attribute__((ext_vector_type(2))) float v2f;
typedef __attribute__((ext_vector_type(8))) float v8f;

#define NTOT   65536     // B*N
#define NGRAPH 16
#define NNODE  4096
#define KC     64
#define DF     128
#define NEDGE  1048576   // B*N*DEG
#define EPER   65536.0f  // edges per graph

// workspace layout (float element offsets)
#define WS_CLUSTER 0        // int[65536]
#define WS_SH      65536    // float[65536]
#define WS_CNT2    131072   // float[1024]  per (graph,cluster) sum of s_h^2
#define WS_CROSS   132096   // float[16]
#define WS_ASQ     132112   // float[16]
#define WS_ENT     132128   // float[1]

// d_out layout (float element offsets), total 197651
#define OUT_POOL   0        // [1024*128]
#define OUT_ADJ    131072   // [16*64*64]
#define OUT_LINK   196608
#define OUT_ENT    196609
#define OUT_BATCH  196610   // [1024]
#define OUT_BPTR   197634   // [17]

// ---------------------------------------------------------------- init ----
__global__ __launch_bounds__(256) void k_init(float* __restrict__ outp,
                                              float* __restrict__ ws) {
    int i = blockIdx.x * 256 + threadIdx.x;
    if (i < NGRAPH * KC * KC) outp[OUT_ADJ + i] = 0.0f;     // atomic target
    if (i < 1057)             ws[WS_CNT2 + i] = 0.0f;       // cnt2+cross+asq+ent
}

// ------------------------------------------------------- assign (WMMA) ----
// One wave = 16 nodes. logits[16x64] = X[16x128] @ W[128x64] + b via
// v_wmma_f32_16x16x4_f32 (4 N-tiles, 32 K-steps). Then gumbel + softmax.
__global__ __launch_bounds__(256) void k_assign(
    const float* __restrict__ x, const float* __restrict__ W,
    const float* __restrict__ bias, const float* __restrict__ gu,
    int* __restrict__ cluster, float* __restrict__ sh,
    float* __restrict__ cnt2, float* __restrict__ entAcc)
{
    __shared__ float zbuf[8 * 16 * 64];   // 8 waves x (16 nodes x 64 clusters)
    __shared__ float cl2[KC];
    __shared__ float red[256];

    const int tid  = threadIdx.x;
    const int wave = tid >> 5, lane = tid & 31;
    const int row  = lane & 15, hi = lane >> 4;
    const int base = blockIdx.x * 128 + wave * 16;   // first node of this wave

    if (tid < KC) cl2[tid] = 0.0f;

    // C tiles init with bias: element (M = v+8*hi, N = row + 16t) -> b[16t+row]
    v8f acc[4];
    #pragma unroll
    for (int t = 0; t < 4; ++t) {
        float bv = bias[t * 16 + row];
        #pragma unroll
        for (int j = 0; j < 8; ++j) acc[t][j] = bv;
    }

    // A tile: lane holds (M=row, K = 4*kk + 2*hi + {0,1})
    const float* xrow = x + (size_t)(base + row) * DF + 2 * hi;
    #pragma unroll 4
    for (int kk = 0; kk < 32; ++kk) {
        const int d0 = 4 * kk + 2 * hi;
        v2f a = *(const v2f*)(xrow + 4 * kk);
        #pragma unroll
        for (int t = 0; t < 4; ++t) {
            v2f bb;   // B tile: (K = 2*hi + {0,1}, N = row) of W[d0.., 16t..]
            bb.x = W[(size_t)d0 * KC + t * 16 + row];
            bb.y = W[(size_t)(d0 + 1) * KC + t * 16 + row];
            acc[t] = __builtin_amdgcn_wmma_f32_16x16x4_f32(
                false, a, false, bb, (short)0, acc[t], false, false);
        }
    }

    // z = logits + gumbel -> LDS (C layout: M = v + 8*hi, N = 16t + row)
    #pragma unroll
    for (int t = 0; t < 4; ++t) {
        #pragma unroll
        for (int v = 0; v < 8; ++v) {
            int M = v + 8 * hi;
            int col = t * 16 + row;
            float u = gu[(size_t)(base + M) * KC + col];
            float g = -logf(-logf(u));
            zbuf[wave * 1024 + M * 64 + col] = acc[t][v] + g;
        }
    }
    __syncthreads();

    // per-node softmax / argmax: threads 0..127 each take one node
    float entv = 0.0f;
    if (tid < 128) {
        const float* zp = &zbuf[(tid >> 4) * 1024 + (tid & 15) * 64];
        float m = zp[0]; int am = 0;
        #pragma unroll 8
        for (int k = 1; k < KC; ++k) { float z = zp[k]; if (z > m) { m = z; am = k; } }
        float ssum = 0.0f;
        #pragma unroll 8
        for (int k = 0; k < KC; ++k) ssum += expf(zp[k] - m);
        float p   = 1.0f / ssum;              // softmax value at argmax
        float s_h = (1.0f + p) - p;           // straight-through value, fp-exact
        int gnode = blockIdx.x * 128 + tid;
        cluster[gnode] = am;
        sh[gnode]      = s_h;
        atomicAdd(&cl2[am], s_h * s_h);       // StS diagonal
        entv = -s_h * logf(s_h + 1e-15f);     // only nonzero entropy term
    }
    red[tid] = entv;
    __syncthreads();
    for (int off = 128; off > 0; off >>= 1) {
        if (tid < off) red[tid] += red[tid + off];
        __syncthreads();
    }
    if (tid == 0) atomicAdd(entAcc, red[0]);
    if (tid < KC) {
        int gb = (blockIdx.x * 128) >> 12;    // graph id (N = 4096)
        atomicAdd(&cnt2[gb * KC + tid], cl2[tid]);
    }
}

// ----------------------------------------------------------- edge pass ----
// out_adj[b,cs,cd] += w*sh_s*sh_d ; cross_b += same when cs==cd ; a_sq += w^2
__global__ __launch_bounds__(256) void k_edges(
    const int* __restrict__ eidx, const float* __restrict__ ew,
    const int* __restrict__ cluster, const float* __restrict__ sh,
    float* __restrict__ out_adj, float* __restrict__ cross, float* __restrict__ asq)
{
    __shared__ float r1[256], r2[256];
    const int tid = threadIdx.x;
    const int e = blockIdx.x * 256 + tid;
    int s = eidx[e];
    int d = eidx[NEDGE + e];
    float w = ew[e];
    int cs = cluster[s], cd = cluster[d];
    float prod = w * sh[s] * sh[d];
    int gb = blockIdx.x >> 8;                 // 256 edge-blocks per graph
    atomicAdd(&out_adj[((size_t)gb * KC + cs) * KC + cd], prod);
    r1[tid] = (cs == cd) ? prod : 0.0f;
    r2[tid] = w * w;
    __syncthreads();
    for (int off = 128; off > 0; off >>= 1) {
        if (tid < off) { r1[tid] += r1[tid + off]; r2[tid] += r2[tid + off]; }
        __syncthreads();
    }
    if (tid == 0) { atomicAdd(&cross[gb], r1[0]); atomicAdd(&asq[gb], r2[0]); }
}

// -------------------------------------------- pooled features (WMMA) ------
// out_b[64x128] = S_b^T (64x4096, implicit one-hot*s_h) @ X_b (4096x128)
// one wave per 16x16 C tile; A synthesized from cluster[]/sh[] on the fly.
__global__ __launch_bounds__(32) void k_pool_gemm(
    const float* __restrict__ x, const int* __restrict__ cluster,
    const float* __restrict__ sh, float* __restrict__ outp)
{
    const int bid = blockIdx.x;         // 16 graphs * 4 k-tiles * 8 d-tiles
    const int b   = bid >> 5;
    const int mt  = (bid >> 3) & 3;
    const int nt  = bid & 7;
    const int lane = threadIdx.x;
    const int row = lane & 15, hi = lane >> 4;
    const int k0 = mt * 16, c0 = nt * 16;
    const int krow = k0 + row;
    const int nb = b * NNODE;

    v8f acc;
    #pragma unroll
    for (int j = 0; j < 8; ++j) acc[j] = 0.0f;

    for (int n0 = 0; n0 < NNODE; n0 += 4) {
        int na = nb + n0 + 2 * hi;
        int c1 = cluster[na], c2 = cluster[na + 1];
        v2f a, bb;
        a.x = (c1 == krow) ? sh[na]     : 0.0f;   // S^T(M=krow, K=na)
        a.y = (c2 == krow) ? sh[na + 1] : 0.0f;
        bb.x = x[(size_t)na * DF + c0 + row];     // X(K=na, N=c0+row)
        bb.y = x[(size_t)(na + 1) * DF + c0 + row];
        acc = __builtin_amdgcn_wmma_f32_16x16x4_f32(
            false, a, false, bb, (short)0, acc, false, false);
    }
    #pragma unroll
    for (int v = 0; v < 8; ++v) {
        int k = k0 + v + 8 * hi;
        outp[((size_t)b * KC + k) * DF + c0 + row] = acc[v];
    }
}

// ------------------------------------------------------------ finalize ----
__global__ __launch_bounds__(1024) void k_finalize(
    const float* __restrict__ cnt2, const float* __restrict__ cross,
    const float* __restrict__ asq, const float* __restrict__ entAcc,
    float* __restrict__ outp)
{
    __shared__ float linkRed[NGRAPH];
    const int tid = threadIdx.x;
    if (tid < NGRAPH) {
        float sst = 0.0f;
        for (int k = 0; k < KC; ++k) { float c = cnt2[tid * KC + k]; sst += c * c; }
        float v = asq[tid] - 2.0f * cross[tid] + sst;
        linkRed[tid] = sqrtf(fmaxf(v, 0.0f)) / EPER;
    }
    outp[OUT_BATCH + tid] = (float)(tid >> 6);           // repeat(arange(16),64)
    if (tid < 17) outp[OUT_BPTR + tid] = (float)(tid * KC);
    __syncthreads();
    if (tid == 0) {
        float s = 0.0f;
        for (int g = 0; g < NGRAPH; ++g) s += linkRed[g];
        outp[OUT_LINK] = s / (float)NGRAPH;
        outp[OUT_ENT]  = entAcc[0] / (float)NTOT;
    }
}

// ---------------------------------------------------------------- host ----
extern "C" void kernel_launch(void* const* d_in, const int* in_sizes, int n_in,
                              void* d_out, int out_size, void* d_ws, size_t ws_size,
                              hipStream_t stream)
{
    const float* x  = (const float*)d_in[0];   // [65536,128]
    const float* W  = (const float*)d_in[1];   // [128,64]
    const float* bv = (const float*)d_in[2];   // [64]
    const float* ew = (const float*)d_in[3];   // [E]
    const float* gu = (const float*)d_in[4];   // [65536,64]
    const int*   ei = (const int*)d_in[5];     // [2,E] int32
    (void)in_sizes; (void)n_in; (void)out_size; (void)ws_size;

    float* outp = (float*)d_out;
    float* ws   = (float*)d_ws;
    int*   cluster = (int*)(ws + WS_CLUSTER);
    float* sh      = ws + WS_SH;
    float* cnt2    = ws + WS_CNT2;
    float* cross   = ws + WS_CROSS;
    float* asq     = ws + WS_ASQ;
    float* ent     = ws + WS_ENT;

    k_init     <<<256, 256, 0, stream>>>(outp, ws);
    k_assign   <<<NTOT / 128, 256, 0, stream>>>(x, W, bv, gu, cluster, sh, cnt2, ent);
    k_edges    <<<NEDGE / 256, 256, 0, stream>>>(ei, ew, cluster, sh,
                                                 outp + OUT_ADJ, cross, asq);
    k_pool_gemm<<<NGRAPH * 4 * 8, 32, 0, stream>>>(x, cluster, sh, outp + OUT_POOL);
    k_finalize <<<1, 1024, 0, stream>>>(cnt2, cross, asq, ent, outp);
}